// VDDecoder_66709432041701
// MI455X (gfx1250) — compile-verified
//
#include <hip/hip_runtime.h>

// Problem constants from the reference (T, B fixed by setup_inputs()).
#define T_STEPS 2048
#define BATCH   8192
#define CHUNK   32          // timesteps staged per LDS buffer
#define BW      32          // block width = one wave32 (workgroup barriers are NOPs)

// ---- gfx1250 feature probes -------------------------------------------------
#if defined(__has_builtin)
#  if __has_builtin(__builtin_amdgcn_global_load_async_to_lds_b128)
#    define HAVE_ASYNC_LDS 1
#  endif
#  if __has_builtin(__builtin_amdgcn_tanhf)
#    define HAVE_TANH 1
#  endif
#endif

#define AS1 __attribute__((address_space(1)))
#define AS3 __attribute__((address_space(3)))

typedef int v4i __attribute__((ext_vector_type(4)));

// ---- fast activations (v_tanh_f32 is new on gfx1250; co-issues with VALU) ---
__device__ __forceinline__ float tanh_fast(float x) {
#ifdef HAVE_TANH
    return __builtin_amdgcn_tanhf(x);
#else
    // tanh(x) = 2/(1+e^{-2x}) - 1, via v_exp_f32 / v_rcp_f32
    float e = __builtin_amdgcn_exp2f(-2.885390081777927f * x);  // -2*log2(e)*x
    return fmaf(2.0f, __builtin_amdgcn_rcpf(1.0f + e), -1.0f);
#endif
}

__device__ __forceinline__ float sigm_fast(float x) {
#ifdef HAVE_TANH
    return fmaf(0.5f, __builtin_amdgcn_tanhf(0.5f * x), 0.5f);
#else
    float e = __builtin_amdgcn_exp2f(-1.4426950408889634f * x); // -log2(e)*x
    return __builtin_amdgcn_rcpf(1.0f + e);
#endif
}

__device__ __forceinline__ void wait_async0() {
#ifdef HAVE_ASYNC_LDS
#  if __has_builtin(__builtin_amdgcn_s_wait_asynccnt)
    __builtin_amdgcn_s_wait_asynccnt(0);
#  else
    asm volatile("s_wait_asynccnt 0" ::: "memory");
#  endif
    asm volatile("" ::: "memory");   // keep LDS reads after the wait
#endif
}

// ---- fused 3-layer variational-dropout LSTM ---------------------------------
// One lane = one batch column; full recurrent state lives in registers.
// x stream is double-buffered global->LDS with the CDNA5 async copy engine.
__global__ __launch_bounds__(BW) void vdlstm3_kernel(
    const float* __restrict__ x,
    const float* __restrict__ W1ih, const float* __restrict__ W1hh,
    const float* __restrict__ B1,   const float* __restrict__ M1,
    const float* __restrict__ W2ih, const float* __restrict__ W2hh,
    const float* __restrict__ B2,   const float* __restrict__ M2,
    const float* __restrict__ W3ih, const float* __restrict__ W3hh,
    const float* __restrict__ B3,   const float* __restrict__ M3,
    float* __restrict__ out)
{
    __shared__ float tile[2][CHUNK][BW];   // 8 KB of the 320 KB WGP LDS

    const int lane = threadIdx.x;
    const int col  = blockIdx.x * BW + lane;

    // ---- hoist the 88 weights; uniform addresses -> scalar loads / SGPRs ----
    float w1ih[8], w1hh[16], bb1[8];
    float w2ih[16], w2hh[16], bb2[8];
    float w3ih[8], w3hh[4], bb3[4];
#pragma unroll
    for (int i = 0; i < 8;  ++i) w1ih[i] = W1ih[i];
#pragma unroll
    for (int i = 0; i < 16; ++i) w1hh[i] = W1hh[i];
#pragma unroll
    for (int i = 0; i < 8;  ++i) bb1[i]  = B1[i];
#pragma unroll
    for (int i = 0; i < 16; ++i) w2ih[i] = W2ih[i];
#pragma unroll
    for (int i = 0; i < 16; ++i) w2hh[i] = W2hh[i];
#pragma unroll
    for (int i = 0; i < 8;  ++i) bb2[i]  = B2[i];
#pragma unroll
    for (int i = 0; i < 8;  ++i) w3ih[i] = W3ih[i];
#pragma unroll
    for (int i = 0; i < 4;  ++i) w3hh[i] = W3hh[i];
#pragma unroll
    for (int i = 0; i < 4;  ++i) bb3[i]  = B3[i];

    const float mm1a = M1[2 * col], mm1b = M1[2 * col + 1];
    const float mm2a = M2[2 * col], mm2b = M2[2 * col + 1];
    const float mm3  = M3[col];

    float h1a = 0.f, h1b = 0.f, c1a = 0.f, c1b = 0.f;
    float h2a = 0.f, h2b = 0.f, c2a = 0.f, c2b = 0.f;
    float h3  = 0.f, c3  = 0.f;

    const float* xbase = x + (size_t)blockIdx.x * BW;

    // Stage CHUNK timesteps (CHUNK x BW floats) of x into tile[buf].
    auto issue = [&](int tc, int buf) {
        const float* src = xbase + (size_t)tc * CHUNK * BATCH;
#ifdef HAVE_ASYNC_LDS
#pragma unroll
        for (int k = 0; k < 8; ++k) {              // 256 float4 slots / 32 lanes
            int q   = lane + BW * k;
            int row = q >> 3;                      // timestep within chunk
            int c4  = (q & 7) << 2;                // float offset within row
            const float* gp = src + (size_t)row * BATCH + c4;
            float*       lp = &tile[buf][row][c4];
            __builtin_amdgcn_global_load_async_to_lds_b128(
                (AS1 v4i*)gp, (AS3 v4i*)lp, 0, 0);
        }
#else
#pragma unroll
        for (int k = 0; k < 8; ++k) {
            int q   = lane + BW * k;
            int row = q >> 3;
            int c4  = (q & 7) << 2;
            const float4 v = *(const float4*)(src + (size_t)row * BATCH + c4);
            *(float4*)&tile[buf][row][c4] = v;
        }
#endif
    };

    issue(0, 0);
    wait_async0();
#ifndef HAVE_ASYNC_LDS
    __syncthreads();
#endif

    for (int tc = 0; tc < T_STEPS / CHUNK; ++tc) {
        const int cur = tc & 1;
        if (tc + 1 < T_STEPS / CHUNK) issue(tc + 1, cur ^ 1);  // prefetch next buffer

#pragma unroll 4
        for (int r = 0; r < CHUNK; ++r) {
            const float xv = tile[cur][r][lane];
            float ga[8];

            // ---- layer 1: in=1, H=2 (gate order i,f,g,o) ----
#pragma unroll
            for (int j = 0; j < 8; ++j)
                ga[j] = fmaf(xv, w1ih[j],
                         fmaf(h1a, w1hh[2 * j],
                          fmaf(h1b, w1hh[2 * j + 1], bb1[j])));
            {
                float i0 = sigm_fast(ga[0]), i1 = sigm_fast(ga[1]);
                float f0 = sigm_fast(ga[2]), f1 = sigm_fast(ga[3]);
                float u0 = tanh_fast(ga[4]), u1 = tanh_fast(ga[5]);
                float o0 = sigm_fast(ga[6]), o1 = sigm_fast(ga[7]);
                c1a = fmaf(f0, c1a, i0 * u0);
                c1b = fmaf(f1, c1b, i1 * u1);
                h1a = o0 * tanh_fast(c1a);
                h1b = o1 * tanh_fast(c1b);
            }
            const float xa = h1a * mm1a, xb = h1b * mm1b;  // locked dropout

            // ---- layer 2: in=2, H=2 ----
#pragma unroll
            for (int j = 0; j < 8; ++j)
                ga[j] = fmaf(xa, w2ih[2 * j],
                         fmaf(xb, w2ih[2 * j + 1],
                          fmaf(h2a, w2hh[2 * j],
                           fmaf(h2b, w2hh[2 * j + 1], bb2[j]))));
            {
                float i0 = sigm_fast(ga[0]), i1 = sigm_fast(ga[1]);
                float f0 = sigm_fast(ga[2]), f1 = sigm_fast(ga[3]);
                float u0 = tanh_fast(ga[4]), u1 = tanh_fast(ga[5]);
                float o0 = sigm_fast(ga[6]), o1 = sigm_fast(ga[7]);
                c2a = fmaf(f0, c2a, i0 * u0);
                c2b = fmaf(f1, c2b, i1 * u1);
                h2a = o0 * tanh_fast(c2a);
                h2b = o1 * tanh_fast(c2b);
            }
            const float ya = h2a * mm2a, yb = h2b * mm2b;  // locked dropout

            // ---- layer 3: in=2, H=1 ----
            float q0 = fmaf(ya, w3ih[0], fmaf(yb, w3ih[1], fmaf(h3, w3hh[0], bb3[0])));
            float q1 = fmaf(ya, w3ih[2], fmaf(yb, w3ih[3], fmaf(h3, w3hh[1], bb3[1])));
            float q2 = fmaf(ya, w3ih[4], fmaf(yb, w3ih[5], fmaf(h3, w3hh[2], bb3[2])));
            float q3 = fmaf(ya, w3ih[6], fmaf(yb, w3ih[7], fmaf(h3, w3hh[3], bb3[3])));
            c3 = fmaf(sigm_fast(q1), c3, sigm_fast(q0) * tanh_fast(q2));
            h3 = sigm_fast(q3) * tanh_fast(c3);

            out[(size_t)(tc * CHUNK + r) * BATCH + col] = h3 * mm3;
        }

        wait_async0();        // next chunk's async copies have landed in LDS
#ifndef HAVE_ASYNC_LDS
        __syncthreads();
#endif
    }
}

extern "C" void kernel_launch(void* const* d_in, const int* in_sizes, int n_in,
                              void* d_out, int out_size, void* d_ws, size_t ws_size,
                              hipStream_t stream) {
    (void)in_sizes; (void)n_in; (void)out_size; (void)d_ws; (void)ws_size;
    const float* x    = (const float*)d_in[0];
    const float* W1ih = (const float*)d_in[1];
    const float* W1hh = (const float*)d_in[2];
    const float* b1   = (const float*)d_in[3];
    const float* m1   = (const float*)d_in[4];
    const float* W2ih = (const float*)d_in[5];
    const float* W2hh = (const float*)d_in[6];
    const float* b2   = (const float*)d_in[7];
    const float* m2   = (const float*)d_in[8];
    const float* W3ih = (const float*)d_in[9];
    const float* W3hh = (const float*)d_in[10];
    const float* b3   = (const float*)d_in[11];
    const float* m3   = (const float*)d_in[12];
    float* out = (float*)d_out;

    dim3 grid(BATCH / BW);   // 256 single-wave workgroups -> max WGP spread
    dim3 block(BW);
    vdlstm3_kernel<<<grid, block, 0, stream>>>(
        x, W1ih, W1hh, b1, m1, W2ih, W2hh, b2, m2, W3ih, W3hh, b3, m3, out);
}